// RarityNetwork_53257594470540
// MI455X (gfx1250) — compile-verified
//
#include <hip/hip_runtime.h>
#include <hip/hip_bf16.h>
#include <stdint.h>

#define EPSF 1e-8f
#define OUTW 240
#define OUTN (OUTW*OUTW)   /* 57600 */
#define NWAVES 8

typedef float v2f __attribute__((ext_vector_type(2)));
typedef float v8f __attribute__((ext_vector_type(8)));
typedef int   v4i __attribute__((vector_size(4*sizeof(int))));
typedef __attribute__((address_space(1))) v4i av1_v4i;
typedef __attribute__((address_space(3))) v4i av3_v4i;

#if __has_builtin(__builtin_amdgcn_global_load_async_to_lds_b128) && __has_builtin(__builtin_amdgcn_s_wait_asynccnt)
#define USE_ASYNC_LDS 1
#else
#define USE_ASYNC_LDS 0
#endif

#if __has_builtin(__builtin_amdgcn_wmma_f32_16x16x4_f32)
#define USE_WMMA 1
#else
#define USE_WMMA 0
#endif

__device__ __forceinline__ void atomAddF(float* p, float v){
  __hip_atomic_fetch_add(p, v, __ATOMIC_RELAXED, __HIP_MEMORY_SCOPE_AGENT);
}

// ------------------------------------------------------------------ zero
__global__ void k_zero(float* __restrict__ p, int n){
  int i = blockIdx.x*blockDim.x + threadIdx.x;
  if(i<n) p[i]=0.f;
}

// ------------------------------------------------------- per-channel min/max
template<int A>
__global__ __launch_bounds__(256) void k_minmax(const float* __restrict__ x,
      float* __restrict__ mn_out, float* __restrict__ R_out,
      unsigned* __restrict__ gmaxu){
  constexpr int K1 = (A>50)?3:((A==28)?2:((A==14)?1:0));
  constexpr int N  = A*A;
  const int bc = blockIdx.x;
  const float* p = x + (size_t)bc*N;
  float mn=3.4e38f, mx=-3.4e38f;
  for(int e=threadIdx.x; e<N; e+=256){
    int row=e/A, col=e-row*A;
    bool border = (row<K1)|(row>=A-K1)|(col<K1)|(col>=A-K1);
    float v = border?0.f:p[e];
    mn=fminf(mn,v); mx=fmaxf(mx,v);
  }
  __shared__ float smn[256], smx[256];
  int t=threadIdx.x;
  smn[t]=mn; smx[t]=mx; __syncthreads();
  for(int s=128;s>0;s>>=1){
    if(t<s){ smn[t]=fminf(smn[t],smn[t+s]); smx[t]=fmaxf(smx[t],smx[t+s]); }
    __syncthreads();
  }
  if(t==0){
    float m0=smn[0], R=smx[0]-m0;
    mn_out[bc]=m0; R_out[bc]=R;
    atomicMax(gmaxu, __float_as_uint(fmaxf(R,0.f)));
  }
}

// --------------------------------------------------------------- main rarity
template<int A>
__global__ __launch_bounds__(256) void k_rarity(const float* __restrict__ x,
      const float* __restrict__ mn_in, const float* __restrict__ R_in,
      const unsigned* __restrict__ gmaxu, float* __restrict__ rsum, int C){
  constexpr int K1 = (A>50)?3:((A==28)?2:((A==14)?1:0));
  constexpr int KB = (A>50)?3:2;
  constexpr int N  = A*A;
  constexpr int CHUNK = 1024;            // 256 threads * 4 floats
  __shared__ unsigned char sidx[N];
  __shared__ unsigned shist[NWAVES][3][16];
  __shared__ float stable[16];
#if USE_ASYNC_LDS
  __shared__ float4 stage[NWAVES][2][32];
#endif
  const int bc=blockIdx.x, img=bc/C;
  const int t=threadIdx.x, lane=t&31, w=t>>5;
  const float* p = x + (size_t)bc*N;
  const float mn   = mn_in[bc];
  const float invR = 256.f/(R_in[bc]+EPSF);
  const float gR   = __uint_as_float(*gmaxu);
  const float gmax = 256.f*gR/(gR+EPSF);
  const float binscale = (gmax>0.f)?(11.f/gmax):0.f;

  for(int i=t;i<NWAVES*3*16;i+=256) ((unsigned*)shist)[i]=0u;
  __syncthreads();

  const int base_lane = w*128 + lane*4;
  const int nchunks = (N+CHUNK-1)/CHUNK;

  auto proc=[&](int e0, float4 xv){
    float vals[4]={xv.x,xv.y,xv.z,xv.w};
    #pragma unroll
    for(int i2=0;i2<4;i2++){
      int e=e0+i2;
      if(e<N){
        int row=e/A, col=e-row*A;
        bool b1=(row<K1)|(row>=A-K1)|(col<K1)|(col>=A-K1);
        float xvv = b1?0.f:vals[i2];
        float v=(xvv-mn)*invR;                       // minmax * 256
        int bi=(int)floorf(v*binscale); bi=bi<0?0:(bi>10?10:bi);
        int ix=(int)(v*(10.f/256.f));   ix=ix<0?0:(ix>10?10:ix);
        sidx[e]=(unsigned char)ix;
        atomicAdd(&shist[w][0][bi],1u);
        atomicAdd(&shist[w][1][ix],1u);
        bool ib2=(row>=KB)&(row<A-KB)&(col>=KB)&(col<A-KB);
        if(ib2) atomicAdd(&shist[w][2][ix],1u);
      }
    }
  };

#if USE_ASYNC_LDS
  auto issue=[&](int c,int buf){
    int e0=c*CHUNK+base_lane; if(e0>N-4)e0=N-4;     // clamp keeps EXEC full
    __builtin_amdgcn_global_load_async_to_lds_b128(
        (av1_v4i*)(uintptr_t)(p+e0),
        (av3_v4i*)(uint32_t)(uintptr_t)(&stage[w][buf][lane]), 0, 0);
  };
  issue(0,0);
  for(int c=0;c<nchunks;c++){
    if(c+1<nchunks){ issue(c+1,(c+1)&1); __builtin_amdgcn_s_wait_asynccnt(1); }
    else           { __builtin_amdgcn_s_wait_asynccnt(0); }
    float4 xv = stage[w][c&1][lane];
    proc(c*CHUNK+base_lane, xv);
  }
#else
  for(int c=0;c<nchunks;c++){
    int e0=c*CHUNK+base_lane;
    int ec=e0>N-4?N-4:e0;
    float4 xv = *reinterpret_cast<const float4*>(p+ec);
    proc(e0,xv);
  }
#endif
  __syncthreads();

  if(t<48){                                   // reduce wave-private histograms
    int which=t>>4, j=t&15;
    unsigned s=0;
    #pragma unroll
    for(int ww=0;ww<NWAVES;ww++) s+=shist[ww][which][j];
    shist[0][which][j]=s;
  }
  __syncthreads();

  if(t==0){                                   // scalar chain on 11 bin values
    const float invN=1.f/(float)N;
    unsigned* cbi=&shist[0][0][0];
    unsigned* cix=&shist[0][1][0];
    unsigned* cin=&shist[0][2][0];
    float lut[11];
    for(int j=0;j<11;j++) lut[j]=-logf((float)cbi[j]*invN + 1e-4f);
    float mn1=3.4e38f,mx1=-3.4e38f,s1=0.f;
    for(int j=0;j<11;j++){ if(cix[j]){mn1=fminf(mn1,lut[j]); mx1=fmaxf(mx1,lut[j]);} s1+=(float)cix[j]*lut[j]; }
    float meand=s1*invN;
    float inv1=1.f/(mx1-mn1+EPSF);
    float w1=((mx1-mn1)-(meand-mn1))*inv1; w1*=w1;            // (max-mean)^2 after minmax
    float d3v[11]; float max3=-3.4e38f,s3=0.f;
    for(int j=0;j<11;j++){
      float d1=(lut[j]-mn1)*inv1; float d2=d1*w1; float d3=d2*d2; d3v[j]=d3;
      if(cix[j]) max3=fmaxf(max3,d3); s3+=(float)cix[j]*d3;
    }
    float w2=(max3 - s3*invN); w2*=w2;
    float mn4=3.4e38f,mx4=-3.4e38f;
    for(int j=0;j<11;j++){ if(cix[j]){ float d4=d3v[j]*w2; mn4=fminf(mn4,d4); mx4=fmaxf(mx4,d4);} }
    float inv4=1.f/(mx4-mn4+EPSF);
    float d5v[11];
    for(int j=0;j<11;j++) d5v[j]=(d3v[j]*w2-mn4)*inv4;
    float maxF=0.f,sF=0.f;                                    // borders contribute 0
    for(int j=0;j<11;j++){ float f=d5v[j]*d5v[j]; if(cin[j]) maxF=fmaxf(maxF,f); sF+=(float)cin[j]*f; }
    float wF=(maxF - sF*invN); wF*=wF;
    for(int j=0;j<11;j++) stable[j]=d5v[j]*d5v[j]*wF;
  }
  __syncthreads();

  float* rdst = rsum + (size_t)img*N;
  for(int e=t;e<N;e+=256){
    int row=e/A, col=e-row*A;
    bool ib2=(row>=KB)&(row<A-KB)&(col>=KB)&(col<A-KB);
    if(ib2) atomAddF(&rdst[e], stable[sidx[e]]);
  }
}

// ---------------------------------------------- bilinear weight matrix (SxOUTW)
__global__ void k_buildW(float* __restrict__ W, int S){
  int i = blockIdx.x*blockDim.x+threadIdx.x;
  if(i>=S*OUTW) return;
  int y=i/OUTW, j=i-y*OUTW;
  float src = (j+0.5f)*(float)S/(float)OUTW - 0.5f;   // half-pixel centers
  float fl = floorf(src);
  int y0=(int)fl; float f=src-fl;
  int y0c=min(max(y0,0),S-1), y1c=min(max(y0+1,0),S-1);
  float wgt=0.f;
  if(y==y0c) wgt+=1.f-f;
  if(y==y1c) wgt+=f;
  W[(size_t)y*OUTW+j]=wgt;
}

// ---------------------------------------------------- WMMA f32 GEMM (16x16 tile/wave)
// C[M,N] = A[M,K] * B[K,N]; TRA reads A transposed.
// Out-of-range rows/cols are handled by ADDRESS CLAMPING (they only feed
// C entries that are never stored), so the hot loop has zero guards.
// Dual accumulators break the WMMA->WMMA RAW dependency (ISA 7.12.1).
template<bool TRA>
__global__ __launch_bounds__(256) void k_gemm(const float* __restrict__ Ap,
      const float* __restrict__ Bp, float* __restrict__ Cp,
      int M,int N,int K,int lda,int ldb,int ldc,
      long sA,long sB,long sC,int tilesM,int tilesN,int nimg){
  int gwid=(blockIdx.x*256+threadIdx.x)>>5;
  int lane=threadIdx.x&31;
  int tpi=tilesM*tilesN;
  int img=gwid/tpi;
  if(img>=nimg) return;                         // wave-uniform: EXEC stays full
  int tl=gwid-img*tpi;
  int tmq=tl/tilesN;
  int tm=tmq*16, tn=(tl-tmq*tilesN)*16;
  const float* Ab=Ap+(size_t)img*sA;
  const float* Bb=Bp+(size_t)img*sB;
  float*       Cb=Cp+(size_t)img*sC;
  int half=lane>>4, lm=lane&15;
  int am=tm+lm, bn=tn+lm;
  int amc = am<M?am:(M-1);                      // address clamp, not zeroing
  int bnc = bn<N?bn:(N-1);
  v8f acc0={0.f,0.f,0.f,0.f,0.f,0.f,0.f,0.f};
  v8f acc1={0.f,0.f,0.f,0.f,0.f,0.f,0.f,0.f};
  const int Kfloor = K & ~3;

#if USE_WMMA
  const float* pB = Bb + (size_t)(2*half)*ldb + bnc;
  if(!TRA){
    const float* pA = Ab + (size_t)amc*lda + 2*half;   // 8B-aligned (lda even)
    int k=0;
    for(; k+8<=Kfloor; k+=8){
      v2f af0 = *reinterpret_cast<const v2f*>(pA);
      v2f bf0; bf0.x=pB[0]; bf0.y=pB[ldb];
      v2f af1 = *reinterpret_cast<const v2f*>(pA+4);
      v2f bf1; bf1.x=pB[(size_t)4*ldb]; bf1.y=pB[(size_t)5*ldb];
      acc0 = __builtin_amdgcn_wmma_f32_16x16x4_f32(false, af0, false, bf0, (short)0, acc0, false, false);
      acc1 = __builtin_amdgcn_wmma_f32_16x16x4_f32(false, af1, false, bf1, (short)0, acc1, false, false);
      pA += 8; pB += (size_t)8*ldb;
    }
    for(; k<Kfloor; k+=4){
      v2f af = *reinterpret_cast<const v2f*>(pA);
      v2f bf; bf.x=pB[0]; bf.y=pB[ldb];
      acc0 = __builtin_amdgcn_wmma_f32_16x16x4_f32(false, af, false, bf, (short)0, acc0, false, false);
      pA += 4; pB += (size_t)4*ldb;
    }
  } else {
    const float* pA = Ab + (size_t)(2*half)*lda + amc;
    int k=0;
    for(; k+8<=Kfloor; k+=8){
      v2f af0; af0.x=pA[0];              af0.y=pA[lda];
      v2f bf0; bf0.x=pB[0];              bf0.y=pB[ldb];
      v2f af1; af1.x=pA[(size_t)4*lda];  af1.y=pA[(size_t)5*lda];
      v2f bf1; bf1.x=pB[(size_t)4*ldb];  bf1.y=pB[(size_t)5*ldb];
      acc0 = __builtin_amdgcn_wmma_f32_16x16x4_f32(false, af0, false, bf0, (short)0, acc0, false, false);
      acc1 = __builtin_amdgcn_wmma_f32_16x16x4_f32(false, af1, false, bf1, (short)0, acc1, false, false);
      pA += (size_t)8*lda; pB += (size_t)8*ldb;
    }
    for(; k<Kfloor; k+=4){
      v2f af; af.x=pA[0]; af.y=pA[lda];
      v2f bf; bf.x=pB[0]; bf.y=pB[ldb];
      acc0 = __builtin_amdgcn_wmma_f32_16x16x4_f32(false, af, false, bf, (short)0, acc0, false, false);
      pA += (size_t)4*lda; pB += (size_t)4*ldb;
    }
  }
  if(Kfloor<K){                                  // K%4 tail (only A==14)
    int k0=Kfloor+2*half;
    v2f af,bf;
    af.x = (k0  <K) ? (TRA?Ab[(size_t)k0*lda+amc]    :Ab[(size_t)amc*lda+k0])   : 0.f;
    af.y = (k0+1<K) ? (TRA?Ab[(size_t)(k0+1)*lda+amc]:Ab[(size_t)amc*lda+k0+1]) : 0.f;
    bf.x = (k0  <K) ? Bb[(size_t)k0*ldb+bnc]     : 0.f;
    bf.y = (k0+1<K) ? Bb[(size_t)(k0+1)*ldb+bnc] : 0.f;
    acc0 = __builtin_amdgcn_wmma_f32_16x16x4_f32(false, af, false, bf, (short)0, acc0, false, false);
  }
  v8f acc = acc0 + acc1;
#else
  (void)amc;(void)bnc;
  v8f acc=acc0;
  for(int k=0;k<K;k+=4){
    #pragma unroll
    for(int r=0;r<8;r++){
      int row=tm+r+8*half;
      float a0=0.f;
      #pragma unroll
      for(int kk=0;kk<4;kk++){
        int kx=k+kk;
        float av=(row<M&&kx<K)?(TRA?Ab[(size_t)kx*lda+row]:Ab[(size_t)row*lda+kx]):0.f;
        float bv=(kx<K&&bn<N)?Bb[(size_t)kx*ldb+bn]:0.f;
        a0+=av*bv;
      }
      acc[r]+=a0;
    }
  }
  (void)acc1;
#endif
  #pragma unroll
  for(int r=0;r<8;r++){
    int row=tm+r+8*half, col=tn+lm;
    if(row<M && col<N) Cb[(size_t)row*ldc+col]=acc[r];
  }
}

// -------------------------------------------------- fuse stage: per-map stats
__global__ __launch_bounds__(256) void k_fstat(const float* __restrict__ lmaps,
                                               float* __restrict__ fstat){
  int mb=blockIdx.x;
  const float* p=lmaps+(size_t)mb*OUTN;
  float mn=3.4e38f,mx=-3.4e38f,s=0.f;
  for(int e=threadIdx.x;e<OUTN;e+=256){ float v=p[e]; mn=fminf(mn,v); mx=fmaxf(mx,v); s+=v; }
  __shared__ float smn[256],smx[256],ss[256];
  int t=threadIdx.x; smn[t]=mn;smx[t]=mx;ss[t]=s; __syncthreads();
  for(int st=128;st>0;st>>=1){
    if(t<st){smn[t]=fminf(smn[t],smn[t+st]);smx[t]=fmaxf(smx[t],smx[t+st]);ss[t]+=ss[t+st];}
    __syncthreads();
  }
  if(t==0){ fstat[mb*3]=smn[0]; fstat[mb*3+1]=smx[0]-smn[0]; fstat[mb*3+2]=ss[0]*(1.f/(float)OUTN); }
}

// -------------------------------------------- fuse stage: weighted group maps
__global__ __launch_bounds__(256) void k_group(const float* __restrict__ lmaps,
      const float* __restrict__ fstat, float* __restrict__ out,
      float* __restrict__ gstat){
  const int gcnt[5]={2,2,3,3,3};
  const int gmem[5][3]={{0,1,0},{2,3,0},{4,5,6},{7,8,9},{10,11,12}};
  int bg=blockIdx.x; int b=bg/5, g=bg-5*b;
  int nc=gcnt[g];
  float scale[3]; float bias=0.f;
  const float* src[3];
  for(int c=0;c<nc;c++){
    int m=gmem[g][c];
    const float* st=&fstat[(m*8+b)*3];
    float mnv=st[0],R=st[1],mean=st[2];
    float inv=1.f/(R+EPSF);
    float wgt=(R-(mean-mnv))*inv; wgt*=wgt;     // (max-mean)^2 of normalized map
    scale[c]=wgt*inv;
    bias -= wgt*inv*mnv;
    src[c]=lmaps+((size_t)m*8+b)*OUTN;
  }
  float* dst = out + (size_t)8*OUTN + (size_t)bg*OUTN;
  float mn=3.4e38f,mx=-3.4e38f;
  for(int e=threadIdx.x;e<OUTN;e+=256){
    float v=bias;
    for(int c=0;c<nc;c++) v+=scale[c]*src[c][e];
    dst[e]=v;
    mn=fminf(mn,v); mx=fmaxf(mx,v);
  }
  __shared__ float smn[256],smx[256];
  int t=threadIdx.x; smn[t]=mn;smx[t]=mx; __syncthreads();
  for(int st=128;st>0;st>>=1){
    if(t<st){smn[t]=fminf(smn[t],smn[t+st]);smx[t]=fmaxf(smx[t],smx[t+st]);}
    __syncthreads();
  }
  if(t==0){ gstat[bg*2]=smn[0]; gstat[bg*2+1]=smx[0]; }
}

// ------------------------------------ final: normalize groups *255 + group sum
__global__ void k_final(float* __restrict__ out, const float* __restrict__ gstat){
  int i=blockIdx.x*blockDim.x+threadIdx.x;
  if(i>=8*OUTN) return;
  int b=i/OUTN, pp=i-b*OUTN;
  float s=0.f;
  for(int g=0;g<5;g++){
    int bg=b*5+g;
    float* q = out + (size_t)8*OUTN + (size_t)bg*OUTN + pp;
    float mnv=gstat[bg*2], mxv=gstat[bg*2+1];
    float v=(*q-mnv)/(mxv-mnv+EPSF)*255.f;
    *q=v; s+=v;
  }
  out[i]=s;
}

// =========================================================== host orchestration
extern "C" void kernel_launch(void* const* d_in, const int* in_sizes, int n_in,
                              void* d_out, int out_size, void* d_ws, size_t ws_size,
                              hipStream_t stream){
  (void)in_sizes; (void)n_in; (void)out_size; (void)ws_size;
  static const int TC[13]={64,64,128,128,256,256,256,512,512,512,512,512,512};
  static const int TAr[13]={224,224,112,112,56,56,56,28,28,28,14,14,14};
  float* ws=(float*)d_ws;
  size_t off=0;
  const size_t gOff=off; off+=16;                              // 13 gmax uints (zeroed)
  size_t rOff[13]; for(int t=0;t<13;t++){ rOff[t]=off; off+=(size_t)8*TAr[t]*TAr[t]; }
  const size_t zeroCount=off;                                  // zero [0, off)
  size_t mnOff[13], ROff[13];
  for(int t=0;t<13;t++){ mnOff[t]=off; off+=(size_t)8*TC[t]; ROff[t]=off; off+=(size_t)8*TC[t]; }
  static const int WSz[5]={224,112,56,28,14};
  size_t wOff[5]; for(int s=0;s<5;s++){ wOff[s]=off; off+=(size_t)WSz[s]*OUTW; }
  const size_t tmpOff=off; off+=(size_t)8*224*OUTW;
  const size_t lOff=off;   off+=(size_t)13*8*OUTN;
  const size_t fOff=off;   off+=(size_t)13*8*3;
  const size_t gsOff=off;  off+=(size_t)8*5*2;

  { int n=(int)zeroCount; k_zero<<<(n+255)/256,256,0,stream>>>(ws,n); }
  for(int s=0;s<5;s++){
    int n=WSz[s]*OUTW;
    k_buildW<<<(n+255)/256,256,0,stream>>>(ws+wOff[s], WSz[s]);
  }

  for(int t=0;t<13;t++){
    const float* X=(const float*)d_in[t];
    const int C=TC[t], A=TAr[t];
    const int blocks=8*C;
    unsigned* gm=(unsigned*)(ws+gOff)+t;
    float* mnp=ws+mnOff[t];
    float* Rp =ws+ROff[t];
    float* rT =ws+rOff[t];

#define LAUNCH_T(AV) do{ \
    k_minmax<AV><<<blocks,256,0,stream>>>(X, mnp, Rp, gm); \
    k_rarity<AV><<<blocks,256,0,stream>>>(X, mnp, Rp, gm, rT, C); }while(0)
    switch(A){
      case 224: LAUNCH_T(224); break;
      case 112: LAUNCH_T(112); break;
      case 56:  LAUNCH_T(56);  break;
      case 28:  LAUNCH_T(28);  break;
      default:  LAUNCH_T(14);  break;
    }
#undef LAUNCH_T

    const int widx=(A==224)?0:(A==112)?1:(A==56)?2:(A==28)?3:4;
    const float* Wm=ws+wOff[widx];
    float* tmp=ws+tmpOff;
    float* lm =ws+lOff + (size_t)t*8*OUTN;
    {   // tmp(A x 240) = r(A x A) * W(A x 240), per image
      int tm=(A+15)/16, tn=OUTW/16;
      int waves=8*tm*tn; int blk=(waves+7)/8;
      k_gemm<false><<<blk,256,0,stream>>>(rT, Wm, tmp, A,OUTW,A, A,OUTW,OUTW,
                                          (long)A*A, 0L, (long)A*OUTW, tm, tn, 8);
    }
    {   // l(240 x 240) = W^T(240 x A) * tmp(A x 240), per image
      int tm=OUTW/16, tn=OUTW/16;
      int waves=8*tm*tn; int blk=(waves+7)/8;
      k_gemm<true><<<blk,256,0,stream>>>(Wm, tmp, lm, OUTW,OUTW,A, OUTW,OUTW,OUTW,
                                         0L, (long)A*OUTW, (long)OUTN, tm, tn, 8);
    }
  }

  float* out=(float*)d_out;
  k_fstat<<<13*8,256,0,stream>>>(ws+lOff, ws+fOff);
  k_group<<<8*5,256,0,stream>>>(ws+lOff, ws+fOff, out, ws+gsOff);
  k_final<<<(8*OUTN+255)/256,256,0,stream>>>(out, ws+gsOff);
}